// QuantumClassifier_57878979281538
// MI455X (gfx1250) — compile-verified
//
#include <hip/hip_runtime.h>
#include <stdint.h>

// out[b*10+q] = cos(2*pi*inputs[b*784+q]/255); weights are dead (phases cancel).
//
// CDNA5 plan: per block, one TDM tensor_load_to_lds gathers a [ROWS x 10] f32
// tile (row stride 784 elements) into contiguous LDS, s_wait_tensorcnt 0,
// barrier, then every thread evaluates cos via v_cos_f32 (revolution-unit
// input: cos(2*pi*x) for input x, so x = pixel/255 needs exactly one VALU mul)
// and writes coalesced b32 stores.

typedef __attribute__((ext_vector_type(4))) unsigned int v4u;
typedef __attribute__((ext_vector_type(4))) int          v4i;
typedef __attribute__((ext_vector_type(8))) int          v8i;

#define QDIM           10
#define ROW_STRIDE     784
#define ROWS_PER_BLOCK 128
#define THREADS        256

__global__ __launch_bounds__(THREADS)
void qc_cos_kernel(const float* __restrict__ in, float* __restrict__ out, int B) {
    __shared__ float tile[ROWS_PER_BLOCK * QDIM];

    const int rowStart = blockIdx.x * ROWS_PER_BLOCK;
    int rowsHere = B - rowStart;
    if (rowsHere > ROWS_PER_BLOCK) rowsHere = ROWS_PER_BLOCK;
    if (rowsHere <= 0) return;

#if __has_builtin(__builtin_amdgcn_tensor_load_to_lds)
    if (threadIdx.x < 32) {  // one wave per block issues the TDM op (EXEC ignored)
        const uint64_t gaddr  = (uint64_t)(uintptr_t)(in + (uint64_t)rowStart * ROW_STRIDE);
        const uint32_t ldsOff = (uint32_t)(uintptr_t)tile;  // flat shared addr: low 32 bits = LDS byte offset

        // ---- D# group 0 (128b): count | lds_addr | global_addr | type=2 ----
        v4u g0;
        g0.x = 1u;                                            // count=1, user descriptor
        g0.y = ldsOff;                                        // lds_addr (bytes)
        g0.z = (uint32_t)gaddr;                               // global_addr[31:0]
        g0.w = (uint32_t)((gaddr >> 32) & 0x1FFFFFFu)         // global_addr[56:32]
             | (2u << 30);                                    // type = 2 ("image")

        // ---- D# group 1 (256b): sizes / strides ----
        v8i g1;
        g1[0] = (int)(2u << 16);                              // data_size=2 (4 bytes), no multicast/pad/iterate
        g1[1] = (int)(((uint32_t)ROW_STRIDE & 0xFFFFu) << 16);// tensor_dim0[15:0]=784 (abar addr = 0)
        g1[2] = (int)(((uint32_t)rowsHere   & 0xFFFFu) << 16);// tensor_dim0[31:16]=0 | tensor_dim1[15:0]=rows
        g1[3] = (int)((uint32_t)QDIM << 16);                  // tensor_dim1[31:16]=0 | tile_dim0=10
        g1[4] = (int)((uint32_t)rowsHere & 0xFFFFu);          // tile_dim1=rows | tile_dim2=0
        g1[5] = ROW_STRIDE;                                   // tensor_dim0_stride[31:0]=784
        g1[6] = 0;                                            // stride hi | tensor_dim1_stride lo
        g1[7] = 0;

        v4i gz = {0, 0, 0, 0};                                // groups 2/3 unused (2D tile)
#if __clang_major__ >= 23
        v8i gz8 = {0, 0, 0, 0, 0, 0, 0, 0};
        __builtin_amdgcn_tensor_load_to_lds(g0, g1, gz, gz, gz8, 0);
#else
        __builtin_amdgcn_tensor_load_to_lds(g0, g1, gz, gz, 0);
#endif
        __builtin_amdgcn_s_wait_tensorcnt(0);                 // TENSORcnt == 0 -> tile resident in LDS
    }
    __syncthreads();

    const float inv255 = 1.0f / 255.0f;
    const int   n      = rowsHere * QDIM;
    const int   obase  = rowStart * QDIM;
    for (int idx = threadIdx.x; idx < n; idx += THREADS) {
        float x = tile[idx];                                  // stride-256 LDS reads: bank-conflict free
        out[obase + idx] = __builtin_amdgcn_cosf(x * inv255); // v_cos_f32: cos(2*pi * x/255)
    }
#else
    // Fallback (no TDM builtin): direct strided global loads.
    const float inv255 = 1.0f / 255.0f;
    const int   n      = rowsHere * QDIM;
    const int   obase  = rowStart * QDIM;
    for (int idx = threadIdx.x; idx < n; idx += THREADS) {
        int r = idx / QDIM;
        int c = idx - r * QDIM;
        float x = in[(uint64_t)(rowStart + r) * ROW_STRIDE + c];
        out[obase + idx] = __builtin_amdgcn_cosf(x * inv255);
    }
#endif
}

extern "C" void kernel_launch(void* const* d_in, const int* in_sizes, int n_in,
                              void* d_out, int out_size, void* d_ws, size_t ws_size,
                              hipStream_t stream) {
    const float* in  = (const float*)d_in[0];   // [B, 784] f32
    // d_in[1] (weights) is mathematically unused: RZ phases cancel in |a|^2-|b|^2.
    float*       out = (float*)d_out;           // [B, 10] f32

    const int B      = in_sizes[0] / ROW_STRIDE;
    const int blocks = (B + ROWS_PER_BLOCK - 1) / ROWS_PER_BLOCK;
    qc_cos_kernel<<<blocks, THREADS, 0, stream>>>(in, out, B);
}